// Compute_all_u_24653112279162
// MI455X (gfx1250) — compile-verified
//
#include <hip/hip_runtime.h>
#include <stdint.h>

// Vector types (clang ext vectors: support .x/.y/.z/.w and [i])
typedef __attribute__((ext_vector_type(4))) float f4;
typedef __attribute__((ext_vector_type(4))) int   i4;
typedef __attribute__((ext_vector_type(4))) unsigned int v4u;
typedef __attribute__((ext_vector_type(8))) int   v8i;
typedef __attribute__((ext_vector_type(4))) int   v4i;

#define BLOCK 256
#define VOX_PER_THREAD 4
#define TILE (BLOCK * VOX_PER_THREAD)   // 1024 voxels per block
#define TILE_FLOATS (TILE * 3)          // 3072 floats = 12 KB staged in LDS

// out[v,k] = C[0,k] + x*C[1,k] + y*C[2,k] + z*C[3,k],  C = coeffs[elem[v]] (4x3 row-major)
__global__ __launch_bounds__(BLOCK) void voxel_affine_kernel(
    const float* __restrict__ coeffs,     // (E, 4, 3) f32  -> 48 B records, 16B aligned
    const float* __restrict__ centroids,  // (N, 3) f32
    const int*   __restrict__ elem,       // (N,)  i32
    float*       __restrict__ out,        // (N, 3) f32
    int n)
{
    __shared__ float s_cent[TILE_FLOATS];

    const int tile_base = blockIdx.x * TILE;

    // ---------------- TDM: DMA this tile's centroids (contiguous 12 KB) into LDS ----------------
    // One wave-level tensor_load_to_lds per block (TDM ignores EXEC; issue from wave 0 only).
    // Issued FIRST so the DMA runs concurrently with the per-lane index/coeff gathers below.
    if (threadIdx.x == 0) {
        uint64_t gaddr   = (uint64_t)(uintptr_t)(centroids + (size_t)tile_base * 3);
        uint32_t lds_off = (uint32_t)(uintptr_t)(&s_cent[0]);   // flat LDS addr low 32 bits = LDS offset
        uint64_t total   = (uint64_t)n * 3;
        uint64_t start   = (uint64_t)tile_base * 3;
        uint32_t rem     = (uint32_t)(total - start);           // elements remaining; OOB reads -> 0

        // ---- D# group 0 (128b): count=1, lds_addr, global_addr[56:0], type=2 ----
        v4u g0;
        g0.x = 1u;                                              // count=1 (valid user descriptor)
        g0.y = lds_off;                                         // lds_addr (bytes)
        g0.z = (uint32_t)(gaddr & 0xFFFFFFFFull);               // global_addr[31:0]
        g0.w = (uint32_t)((gaddr >> 32) & 0x1FFFFFFull)         // global_addr[56:32]
             | (2u << 30);                                      // type = 2 ("image")

        // ---- D# group 1 (256b): data_size=4B, tensor_dim0=rem, tile_dim0=3072, stride0 ----
        const uint32_t dim0  = rem;          // tensor_dim0 (bits 79:48)
        const uint32_t dim1  = 1u;           // tensor_dim1 (bits 111:80)
        const uint32_t tile0 = TILE_FLOATS;  // tile_dim0   (bits 127:112)
        const uint64_t str0  = TILE_FLOATS;  // tensor_dim0_stride (bits 207:160)

        v8i g1;
        g1[0] = (int)(2u << 16);                                   // workgroup_mask=0, data_size=2 (4B)
        g1[1] = (int)((dim0 & 0xFFFFu) << 16);                     // barrier_addr=0 | dim0[15:0]
        g1[2] = (int)((dim0 >> 16) | ((dim1 & 0xFFFFu) << 16));    // dim0[31:16] | dim1[15:0]
        g1[3] = (int)((dim1 >> 16) | (tile0 << 16));               // dim1[31:16] | tile_dim0
        g1[4] = 0;                                                 // tile_dim1=0, tile_dim2=0 (1D tile)
        g1[5] = (int)(uint32_t)(str0 & 0xFFFFFFFFull);             // stride0[31:0]
        g1[6] = (int)(uint32_t)((str0 >> 32) & 0xFFFFull);         // stride0[47:32] | stride1[15:0]=0
        g1[7] = 0;                                                 // stride1[47:16]=0

        v4i g2 = {0, 0, 0, 0};       // groups 2/3 unused for <=2D tensors
        v4i g3 = {0, 0, 0, 0};
        v8i g4 = {0, 0, 0, 0, 0, 0, 0, 0};  // extra group (6-arg toolchain variant), unused for 1D

        __builtin_amdgcn_tensor_load_to_lds(g0, g1, g2, g3, g4, /*cpol=*/0);
    }

    // ---------------- Overlap phase: issue the latency-critical gathers while TDM runs -------
    // Indices and coeff blocks do not depend on LDS contents, so fetch them BEFORE the barrier.
    const int t    = threadIdx.x;
    const int v0   = tile_base + t * VOX_PER_THREAD;
    const bool full = (v0 + VOX_PER_THREAD <= n);

    f4 ca[VOX_PER_THREAD], cb[VOX_PER_THREAD], cc[VOX_PER_THREAD];
    if (full) {
        i4 idx = *(const i4*)(elem + v0);           // b128 index load (streamed once)
#pragma unroll
        for (int j = 0; j < VOX_PER_THREAD; ++j) {
            const f4* cp = (const f4*)(coeffs + (size_t)idx[j] * 12);  // 48 B record, L2-resident
            ca[j] = cp[0];                          // C[0..3]
            cb[j] = cp[1];                          // C[4..7]
            cc[j] = cp[2];                          // C[8..11]
        }
    }

    // Wave 0 drains the TDM, then everyone syncs: LDS centroids are now valid.
    if (threadIdx.x == 0) {
        __builtin_amdgcn_s_wait_tensorcnt(0);       // s_wait_tensorcnt 0
    }
    __syncthreads();

    if (v0 >= n) return;

    // 12 contiguous centroid floats for this thread's 4 voxels (LDS, 16B aligned)
    float cf[12];
    {
        const f4* sc = (const f4*)(s_cent + t * 12);
        *(f4*)(cf + 0) = sc[0];
        *(f4*)(cf + 4) = sc[1];
        *(f4*)(cf + 8) = sc[2];
    }

    if (full) {
        float r[12];
#pragma unroll
        for (int j = 0; j < VOX_PER_THREAD; ++j) {
            f4 a = ca[j], b = cb[j], c = cc[j];
            float x = cf[3 * j + 0], y = cf[3 * j + 1], z = cf[3 * j + 2];
            r[3 * j + 0] = fmaf(z, c.y, fmaf(y, b.z, fmaf(x, a.w, a.x)));
            r[3 * j + 1] = fmaf(z, c.z, fmaf(y, b.w, fmaf(x, b.x, a.y)));
            r[3 * j + 2] = fmaf(z, c.w, fmaf(y, c.x, fmaf(x, b.y, a.z)));
        }
        // 48 contiguous output bytes per thread; NT stores keep the 96 MB write
        // stream from evicting the L2-resident 24 MB coeff table.
        f4* op = (f4*)(out + (size_t)v0 * 3);
        __builtin_nontemporal_store(*(f4*)(r + 0), op + 0);
        __builtin_nontemporal_store(*(f4*)(r + 4), op + 1);
        __builtin_nontemporal_store(*(f4*)(r + 8), op + 2);
    } else {
        // Tail path (last partial tile only): scalar, bounds-checked
        for (int j = 0; j < VOX_PER_THREAD; ++j) {
            int v = v0 + j;
            if (v >= n) break;
            int e = elem[v];
            const float* C = coeffs + (size_t)e * 12;
            float x = cf[3 * j + 0], y = cf[3 * j + 1], z = cf[3 * j + 2];
            float u0 = fmaf(z, C[9],  fmaf(y, C[6], fmaf(x, C[3], C[0])));
            float u1 = fmaf(z, C[10], fmaf(y, C[7], fmaf(x, C[4], C[1])));
            float u2 = fmaf(z, C[11], fmaf(y, C[8], fmaf(x, C[5], C[2])));
            out[(size_t)v * 3 + 0] = u0;
            out[(size_t)v * 3 + 1] = u1;
            out[(size_t)v * 3 + 2] = u2;
        }
    }
}

extern "C" void kernel_launch(void* const* d_in, const int* in_sizes, int n_in,
                              void* d_out, int out_size, void* d_ws, size_t ws_size,
                              hipStream_t stream) {
    const float* coeffs    = (const float*)d_in[0];  // (E,4,3) f32
    const float* centroids = (const float*)d_in[1];  // (N,3)   f32
    const int*   elem      = (const int*)d_in[2];    // (N,)    i32
    float* out = (float*)d_out;                      // (N,3)   f32

    const int n = in_sizes[2];                       // N voxels
    const int blocks = (n + TILE - 1) / TILE;

    voxel_affine_kernel<<<blocks, BLOCK, 0, stream>>>(coeffs, centroids, elem, out, n);
}